// KFA_88038239633557
// MI455X (gfx1250) — compile-verified
//
#include <hip/hip_runtime.h>

typedef __attribute__((ext_vector_type(2))) float v2f;
typedef __attribute__((ext_vector_type(8))) float v8f;

#define N_PTS 4096
#define N_CH  64
#define N_B   4
#define K_NN  16
#define OUT_H 64

// ---------------------------------------------------------------------------
// Fused Gram-matrix (fp32 WMMA) + streaming top-16 per row.
// One wave (32 lanes) per workgroup handles 32 rows of one (batch, modality).
// pairwise(i,j) = -xx_i - xx_j + 2*dot(x_i,x_j); the -xx_i row constant does
// not change per-row ordering, so score(j) = 2*dot - xx_j is used.
// ---------------------------------------------------------------------------
__global__ void __launch_bounds__(32)
knn_topk_kernel(const float* __restrict__ depth,
                const float* __restrict__ color,
                int* __restrict__ idx_d,
                int* __restrict__ idx_c)
{
    const int lane = threadIdx.x;      // 0..31
    const int half = lane >> 4;        // 0/1 (WMMA K-half)
    const int l16  = lane & 15;        // M / N index within tile
    const int b    = blockIdx.y;
    const int mod  = blockIdx.z;
    const int r0   = blockIdx.x * 32;

    const float* X = (mod == 0 ? depth : color) + (size_t)b * N_CH * N_PTS;
    int* idx_out   = (mod == 0 ? idx_d : idx_c) + ((size_t)b * N_PTS + r0) * K_NN;

    // Preload A fragments: rows r0..r0+31 (two 16-row tiles), all 64 channels.
    // fp32 16x4 A layout: lanes 0-15 hold K=2*half+{0,1} in vgpr0/1, M=lane&15.
    v2f afrag[2][16];
    #pragma unroll
    for (int t = 0; t < 2; ++t) {
        const int row = r0 + t * 16 + l16;
        #pragma unroll
        for (int kc = 0; kc < 16; ++kc) {
            const int ch = kc * 4 + half * 2;
            afrag[t][kc].x = X[(size_t)ch * N_PTS + row];
            afrag[t][kc].y = X[(size_t)(ch + 1) * N_PTS + row];
        }
    }

    float topv[K_NN];
    int   topi[K_NN];
    #pragma unroll
    for (int i = 0; i < K_NN; ++i) { topv[i] = -3.0e38f; topi[i] = 0; }

    __shared__ float tile[32 * 17];    // 32 rows x 16 cols, padded stride 17

    for (int c0 = 0; c0 < N_PTS; c0 += 16) {
        const int col = c0 + l16;

        // B fragments (same fp32 layout, N=col) + partial column norm.
        v2f   bfrag[16];
        float xxp = 0.f;
        #pragma unroll
        for (int kc = 0; kc < 16; ++kc) {
            const int ch = kc * 4 + half * 2;
            const float b0 = X[(size_t)ch * N_PTS + col];
            const float b1 = X[(size_t)(ch + 1) * N_PTS + col];
            bfrag[kc].x = b0;
            bfrag[kc].y = b1;
            xxp = fmaf(b0, b0, xxp);
            xxp = fmaf(b1, b1, xxp);
        }
        // lanes l and l^16 hold complementary channel halves of the same col
        const float xxc = xxp + __shfl_xor(xxp, 16, 32);

        v8f acc0 = {};
        v8f acc1 = {};
        #pragma unroll
        for (int kc = 0; kc < 16; ++kc) {
            acc0 = __builtin_amdgcn_wmma_f32_16x16x4_f32(
                false, afrag[0][kc], false, bfrag[kc], (short)0, acc0, false, false);
            acc1 = __builtin_amdgcn_wmma_f32_16x16x4_f32(
                false, afrag[1][kc], false, bfrag[kc], (short)0, acc1, false, false);
        }

        // C layout: vgpr v, lane -> (M = v + 8*half, N = l16). Transpose via LDS.
        #pragma unroll
        for (int v = 0; v < 8; ++v) {
            const int row = v + 8 * half;
            tile[row * 17 + l16]        = 2.f * acc0[v] - xxc;
            tile[(row + 16) * 17 + l16] = 2.f * acc1[v] - xxc;
        }
        __syncthreads();   // single wave: lowers to a wait, keeps DS ordering explicit

        // Lane owns row r0+lane: stream 16 candidates into sorted top-16.
        #pragma unroll
        for (int j = 0; j < 16; ++j) {
            const float s = tile[lane * 17 + j];
            if (s > topv[K_NN - 1]) {
                topv[K_NN - 1] = s;
                topi[K_NN - 1] = c0 + j;
                #pragma unroll
                for (int p = K_NN - 1; p > 0; --p) {
                    if (topv[p] > topv[p - 1]) {   // strict: stable ties (lower idx first)
                        const float tv = topv[p]; topv[p] = topv[p - 1]; topv[p - 1] = tv;
                        const int   ti = topi[p]; topi[p] = topi[p - 1]; topi[p - 1] = ti;
                    }
                }
            }
        }
        __syncthreads();
    }

    #pragma unroll
    for (int i = 0; i < K_NN; ++i)
        idx_out[lane * K_NN + i] = topi[i];
}

// ---------------------------------------------------------------------------
// fuse = (idx_d + idx_c) as float; out[b,n,o] = sum_k fuse[k]*W[o,k] + bias[o]
// ---------------------------------------------------------------------------
__global__ void __launch_bounds__(256)
fuse_linear_kernel(const int* __restrict__ idx_d,
                   const int* __restrict__ idx_c,
                   const float* __restrict__ W,
                   const float* __restrict__ bias,
                   float* __restrict__ out)
{
    __shared__ float sW[OUT_H * K_NN];
    __shared__ float sB[OUT_H];
    for (int i = threadIdx.x; i < OUT_H * K_NN; i += blockDim.x) sW[i] = W[i];
    for (int i = threadIdx.x; i < OUT_H; i += blockDim.x)       sB[i] = bias[i];
    __syncthreads();

    const int t = blockIdx.x * blockDim.x + threadIdx.x;  // point id in [0, B*N)
    if (t >= N_B * N_PTS) return;

    float f[K_NN];
    #pragma unroll
    for (int k = 0; k < K_NN; ++k)
        f[k] = (float)(idx_d[(size_t)t * K_NN + k] + idx_c[(size_t)t * K_NN + k]);

    float* o = out + (size_t)t * OUT_H;
    #pragma unroll 4
    for (int oi = 0; oi < OUT_H; ++oi) {
        float acc = sB[oi];
        #pragma unroll
        for (int k = 0; k < K_NN; ++k)
            acc = fmaf(f[k], sW[oi * K_NN + k], acc);
        o[oi] = acc;
    }
}

// ---------------------------------------------------------------------------
// Inputs (setup_inputs order): 0=k(scalar), 1=depth, 2=color, 3=W, 4=b
// Workspace: idx_d [B*N*16] ints, idx_c [B*N*16] ints  (~2 MB total)
// ---------------------------------------------------------------------------
extern "C" void kernel_launch(void* const* d_in, const int* in_sizes, int n_in,
                              void* d_out, int out_size, void* d_ws, size_t ws_size,
                              hipStream_t stream)
{
    const float* depth = (const float*)d_in[1];
    const float* color = (const float*)d_in[2];
    const float* W     = (const float*)d_in[3];
    const float* bias  = (const float*)d_in[4];
    float*       out   = (float*)d_out;

    int* idx_d = (int*)d_ws;
    int* idx_c = idx_d + (size_t)N_B * N_PTS * K_NN;

    dim3 g1(N_PTS / 32, N_B, 2);
    knn_topk_kernel<<<g1, 32, 0, stream>>>(depth, color, idx_d, idx_c);

    const int total = N_B * N_PTS;
    fuse_linear_kernel<<<(total + 255) / 256, 256, 0, stream>>>(idx_d, idx_c, W, bias, out);
}